// S2SLoss_70781061038709
// MI455X (gfx1250) — compile-verified
//
#include <hip/hip_runtime.h>
#include <hip/hip_bf16.h>
#include <math.h>

typedef __attribute__((ext_vector_type(2))) float v2f;
typedef __attribute__((ext_vector_type(8))) float v8f;

// ---------------- wave32 butterfly reductions (deterministic) ----------------
__device__ __forceinline__ float wredSum(float v) {
#pragma unroll
  for (int o = 16; o > 0; o >>= 1) v += __shfl_xor(v, o, 32);
  return v;
}
__device__ __forceinline__ float wredMax(float v) {
#pragma unroll
  for (int o = 16; o > 0; o >>= 1) v = fmaxf(v, __shfl_xor(v, o, 32));
  return v;
}
__device__ __forceinline__ int wredSumI(int v) {
#pragma unroll
  for (int o = 16; o > 0; o >>= 1) v += __shfl_xor(v, o, 32);
  return v;
}

// ================= Kernel 1: cross-entropy partials =================
// One wave per row (128 classes -> float4 per lane). 8 rows per block.
__global__ void __launch_bounds__(256)
ce_partial_kernel(const float* __restrict__ pred, const int* __restrict__ labels,
                  float* __restrict__ partial) {
  __shared__ float acc[8];
  const int wave = threadIdx.x >> 5;
  const int lane = threadIdx.x & 31;
  const int row = blockIdx.x * 8 + wave;

  const float4 p = ((const float4*)(pred + (size_t)row * 128))[lane];
  float m = fmaxf(fmaxf(p.x, p.y), fmaxf(p.z, p.w));
  m = wredMax(m);
  float e = expf(p.x - m) + expf(p.y - m) + expf(p.z - m) + expf(p.w - m);
  const float s = wredSum(e);

  const int lbl = labels[row];
  float pl = 0.f;
  if ((lbl >> 2) == lane) {
    switch (lbl & 3) {
      case 0: pl = p.x; break;
      case 1: pl = p.y; break;
      case 2: pl = p.z; break;
      default: pl = p.w; break;
    }
  }
  pl = wredSum(pl);  // broadcast label logit to all lanes
  const float loss = -(pl - m - logf(s));

  if (lane == 0) acc[wave] = loss;
  __syncthreads();
  if (threadIdx.x == 0) {
    float t = 0.f;
#pragma unroll
    for (int i = 0; i < 8; ++i) t += acc[i];
    partial[blockIdx.x] = t;
  }
}

// ================= Kernel 2: physics-score partials =================
// One wave per window [512 x 3]. Window staged to LDS via async copy
// (global_load_async_to_lds_b128 + s_wait_asynccnt), then multi-pass stats.
__global__ void __launch_bounds__(256)
phys_partial_kernel(const float* __restrict__ w, float* __restrict__ partial) {
  __shared__ float buf[8][1536];  // 48 KB
  __shared__ float pens[8];
  const int wave = threadIdx.x >> 5;
  const int lane = threadIdx.x & 31;
  const int win = blockIdx.x * 8 + wave;
  const float* gw = w + (size_t)win * 1536;
  float* Wb = &buf[wave][0];

  // --- async copy global -> LDS: 384 x 16B per wave, 12 x b128 per lane ---
  // ISA 10.2: generic LDS address low 32 bits == LDS byte offset, so we can
  // form the VDST (LDS address) operand from the generic pointer directly.
#pragma unroll
  for (int i = 0; i < 12; ++i) {
    const int idx = i * 32 + lane;  // float4 index within window
    unsigned lds_addr = (unsigned)(size_t)(Wb + idx * 4);
    const float* ga = gw + idx * 4;
    asm volatile("global_load_async_to_lds_b128 %0, %1, off"
                 :
                 : "v"(lds_addr), "v"(ga)
                 : "memory");
  }
  asm volatile("s_wait_asynccnt 0x0" ::: "memory");

  // --- pass 1: per-channel sum and max|w| (LDS stride-3 is bank-conflict-free)
  float s0 = 0.f, s1 = 0.f, s2 = 0.f, m0 = 0.f, m1 = 0.f, m2 = 0.f;
#pragma unroll 4
  for (int t = lane; t < 512; t += 32) {
    const float a = Wb[t * 3 + 0], b = Wb[t * 3 + 1], c = Wb[t * 3 + 2];
    s0 += a; s1 += b; s2 += c;
    m0 = fmaxf(m0, fabsf(a)); m1 = fmaxf(m1, fabsf(b)); m2 = fmaxf(m2, fabsf(c));
  }
  s0 = wredSum(s0); s1 = wredSum(s1); s2 = wredSum(s2);
  m0 = wredMax(m0); m1 = wredMax(m1); m2 = wredMax(m2);
  const float inv_n = 1.0f / 512.0f;
  const float mean0 = s0 * inv_n, mean1 = s1 * inv_n, mean2 = s2 * inv_n;

  // --- pass 2: tv, at-max counts, zero crossings (channel 0) ---
  float tvp = 0.f;
  int c0 = 0, c1 = 0, c2 = 0, zc = 0;
#pragma unroll 2
  for (int t = lane; t < 512; t += 32) {
    const float a = Wb[t * 3 + 0], b = Wb[t * 3 + 1], c = Wb[t * 3 + 2];
    const float d0 = a - mean0, d1 = b - mean1, d2 = c - mean2;
    tvp += d0 * d0 + d1 * d1 + d2 * d2;
    c0 += (fabsf(fabsf(a) - m0) < 0.01f) ? 1 : 0;
    c1 += (fabsf(fabsf(b) - m1) < 0.01f) ? 1 : 0;
    c2 += (fabsf(fabsf(c) - m2) < 0.01f) ? 1 : 0;
    if (t >= 1) {
      const float dp = Wb[(t - 1) * 3] - mean0;
      zc += (d0 * dp < 0.f) ? 1 : 0;
    }
  }

  // --- jerk stencil: jerk[i] = (w[i+4]-2w[i+2]+w[i]) / (4*dt^2), dt=0.02
  // p95 test: sorted[1447] > 5000  <=>  count(|jerk| <= 5000) <= 1447
  int jle = 0;
#pragma unroll 2
  for (int i = lane; i < 508; i += 32) {
#pragma unroll
    for (int ch = 0; ch < 3; ++ch) {
      const float j =
          (Wb[(i + 4) * 3 + ch] - 2.0f * Wb[(i + 2) * 3 + ch] + Wb[i * 3 + ch]) * 625.0f;
      jle += (fabsf(j) <= 5000.0f) ? 1 : 0;
    }
  }

  tvp = wredSum(tvp);
  c0 = wredSumI(c0); c1 = wredSumI(c1); c2 = wredSumI(c2);
  zc = wredSumI(zc); jle = wredSumI(jle);

  const float tv = tvp * inv_n;
  bool reject = (tv < 0.005f);
  reject = reject || ((m0 >= 0.5f) && ((float)c0 * inv_n > 0.2f));
  reject = reject || ((m1 >= 0.5f) && ((float)c1 * inv_n > 0.2f));
  reject = reject || ((m2 >= 0.5f) && ((float)c2 * inv_n > 0.2f));
  reject = reject || (jle <= 1447);

  const float s1v = tv > 10.0f ? 90.0f : (tv > 2.0f ? 80.0f : (tv > 0.5f ? 70.0f : 58.0f));
  const float zcr = (float)zc * inv_n;
  const float s2v = zcr > 0.15f ? 85.0f : (zcr > 0.07f ? 70.0f : 55.0f);
  const float score = floorf((s1v + s2v) * 0.5f);
  const float pen = reject ? 1.0f : (1.0f - score * 0.01f);

  if (lane == 0) pens[wave] = pen;
  __syncthreads();
  if (threadIdx.x == 0) {
    float t = 0.f;
#pragma unroll
    for (int i = 0; i < 8; ++i) t += pens[i];
    partial[blockIdx.x] = t;
  }
}

// ================= Kernel 3: WMMA final reduction =================
// Sum via D = ones(16x4) x B(4x16) + C. A is exactly 2 VGPR x 32 lanes = the
// whole 16x4 matrix, so all-ones A makes sum(all D slots) = 16 * sum(B data)
// regardless of the striping details. Single wave, EXEC all ones.
__global__ void __launch_bounds__(32)
final_reduce_kernel(const float* __restrict__ pA, const float* __restrict__ pB,
                    float* __restrict__ out, int nChunks, float invCount) {
  const int lane = threadIdx.x;  // 0..31
  v2f ones; ones[0] = 1.0f; ones[1] = 1.0f;
  v8f accA = {0.f, 0.f, 0.f, 0.f, 0.f, 0.f, 0.f, 0.f};
  v8f accB = {0.f, 0.f, 0.f, 0.f, 0.f, 0.f, 0.f, 0.f};
  for (int ch = 0; ch < nChunks; ++ch) {
    v2f a, b;
    a[0] = pA[ch * 64 + lane];
    a[1] = pA[ch * 64 + 32 + lane];
    b[0] = pB[ch * 64 + lane];
    b[1] = pB[ch * 64 + 32 + lane];
    accA = __builtin_amdgcn_wmma_f32_16x16x4_f32(false, ones, false, a, (short)0, accA,
                                                 false, false);
    accB = __builtin_amdgcn_wmma_f32_16x16x4_f32(false, ones, false, b, (short)0, accB,
                                                 false, false);
  }
  float sA = 0.f, sB = 0.f;
#pragma unroll
  for (int i = 0; i < 8; ++i) { sA += accA[i]; sB += accB[i]; }
  sA = wredSum(sA);
  sB = wredSum(sB);
  if (lane == 0) {
    const float taskLoss = sA * (1.0f / 16.0f) * invCount;  // mean over B rows
    const float physPen = sB * (1.0f / 16.0f) * invCount;   // mean over B windows
    out[0] = taskLoss + 0.3f * physPen;
  }
}

// ================= launcher =================
extern "C" void kernel_launch(void* const* d_in, const int* in_sizes, int n_in,
                              void* d_out, int out_size, void* d_ws, size_t ws_size,
                              hipStream_t stream) {
  const float* pred = (const float*)d_in[0];   // [32768,128] f32
  const int* labels = (const int*)d_in[1];     // [32768] int
  const float* imu = (const float*)d_in[2];    // [32768,512,3] f32
  float* out = (float*)d_out;

  const int B = in_sizes[1];        // 32768 rows / windows
  const int nblocks = B / 8;        // 4096 partials per reduction
  float* wsA = (float*)d_ws;        // [nblocks] CE partials
  float* wsB = wsA + nblocks;       // [nblocks] physics partials

  ce_partial_kernel<<<nblocks, 256, 0, stream>>>(pred, labels, wsA);
  phys_partial_kernel<<<nblocks, 256, 0, stream>>>(imu, wsB);
  final_reduce_kernel<<<1, 32, 0, stream>>>(wsA, wsB, out, nblocks / 64,
                                            1.0f / (float)B);
}